// PBCAgnosticDirectElectrostaticEnergyBlock_53506702574012
// MI455X (gfx1250) — compile-verified
//
#include <hip/hip_runtime.h>
#include <hip/hip_bf16.h>

typedef __attribute__((ext_vector_type(2))) float v2f;
typedef __attribute__((ext_vector_type(8))) float v8f;

// ---------------------------------------------------------------------------
// E[g] = 0.5 * sum_{i,j in g, i != j} q_i * q_j / |p_i - p_j|
//
// 16x16 pair tiles: dist2 = n2_i + n2_j - 2 * (p_i . p_j). The Gram tile
// (p_i . p_j) over 4-padded f32 position vectors is one
// V_WMMA_F32_16X16X4_F32. batch is sorted -> scalar tile-range test skips
// disjoint-graph tiles (~98% of the 512x512 tile grid).
// Inner loop is fully branchless: rsq is unconditional, masking is cndmask.
// ---------------------------------------------------------------------------

__global__ void zero_out_kernel(float* __restrict__ out, int g) {
  int i = blockIdx.x * blockDim.x + threadIdx.x;
  if (i < g) out[i] = 0.0f;
}

__global__ __launch_bounds__(256)
void electro_wmma_kernel(const float* __restrict__ chg,      // [n, kch], q = chg[:,0]
                         const float* __restrict__ pos,      // [n, 3]
                         const long long* __restrict__ batch,// [n], sorted
                         float* __restrict__ out,            // [g]
                         int n, int kch) {
  const int lane  = threadIdx.x & 31;
  const int wave  = threadIdx.x >> 5;
  const int col   = lane & 15;
  const bool hi   = lane >= 16;
  const int rbase = hi ? 8 : 0;          // rows this lane's C-VGPRs cover

  const int ntiles = n >> 4;
  const int iblock = blockIdx.x * 8 + wave;
  if (iblock >= ntiles) return;
  const int ibase = iblock << 4;

  // ---- A operand: i-tile positions, 32-bit A layout (16x4, K in VGPR pairs)
  const int ia   = ibase + col;
  const float px = pos[ia * 3 + 0];
  const float py = pos[ia * 3 + 1];
  const float pz = pos[ia * 3 + 2];
  v2f A;
  A.x = hi ? pz : px;                    // lanes 0-15: {x,y}; lanes 16-31: {z,0}
  A.y = hi ? 0.0f : py;
  const float n2i = px * px + py * py + pz * pz;
  const float qi  = chg[ia * kch];       // q = channel 0
  const int   bi  = (int)batch[ia];

  // Broadcast loop-invariant row-side data (row m lives in lane m's column slot)
  float n2row[8], qrow[8];
  int   brow[8], irow[8];
#pragma unroll
  for (int r = 0; r < 8; ++r) {
    n2row[r] = __shfl(n2i, rbase + r, 32);
    qrow[r]  = __shfl(qi,  rbase + r, 32);
    brow[r]  = __shfl(bi,  rbase + r, 32);
    irow[r]  = ibase + rbase + r;
  }
  const int birmin = __builtin_amdgcn_readfirstlane((int)batch[ibase]);
  const int birmax = __builtin_amdgcn_readfirstlane((int)batch[ibase + 15]);

  float acc[8];
#pragma unroll
  for (int r = 0; r < 8; ++r) acc[r] = 0.0f;

  for (int jt = 0; jt < ntiles; ++jt) {
    const int jbase = jt << 4;
    // batch sorted: scalar (SGPR) range test keeps EXEC full around the WMMA
    const int bjmin = __builtin_amdgcn_readfirstlane((int)batch[jbase]);
    const int bjmax = __builtin_amdgcn_readfirstlane((int)batch[jbase + 15]);
    if (bjmax < birmin || bjmin > birmax) continue;

    // Speculative prefetch of the following tile (global_prefetch_b8)
    if (jbase + 16 < n)
      __builtin_prefetch(&pos[(jbase + 16) * 3 + col], 0, 1);

    // ---- B operand: j-tile positions (same per-lane layout as A)
    const int j    = jbase + col;
    const float jx = pos[j * 3 + 0];
    const float jy = pos[j * 3 + 1];
    const float jz = pos[j * 3 + 2];
    v2f B;
    B.x = hi ? jz : jx;
    B.y = hi ? 0.0f : jy;
    const float n2j = jx * jx + jy * jy + jz * jz;
    const float qj  = chg[j * kch];
    const int   bj  = (int)batch[j];

    // 16x16 tile of p_i . p_j in one matrix op
    v8f c = {0.f, 0.f, 0.f, 0.f, 0.f, 0.f, 0.f, 0.f};
    v8f d = __builtin_amdgcn_wmma_f32_16x16x4_f32(
        /*neg_a=*/false, A, /*neg_b=*/false, B,
        /*c_mod=*/(short)0, c, /*reuse_a=*/false, /*reuse_b=*/false);

    // Branchless per-row epilogue: unconditional rsq, cndmask-only masking.
#pragma unroll
    for (int r = 0; r < 8; ++r) {
      const bool valid = (brow[r] == bj) & (irow[r] != j);
      const float w    = valid ? (qrow[r] * qj) : 0.0f;        // v_cndmask
      float dist2 = n2row[r] + n2j;
      dist2 = __builtin_fmaf(-2.0f, d[r], dist2);              // v_fmac
      dist2 = fmaxf(dist2, 1e-12f);                            // clamp diag/cancel
      const float inv = __builtin_amdgcn_rsqf(dist2);          // v_rsq_f32
      acc[r] = __builtin_fmaf(w, inv, acc[r]);                 // v_fmac
    }
  }

  // Reduce each row across its 16 contributing lanes (columns)
#pragma unroll
  for (int r = 0; r < 8; ++r) {
    float v = acc[r];
    v += __shfl_xor(v, 1, 32);
    v += __shfl_xor(v, 2, 32);
    v += __shfl_xor(v, 4, 32);
    v += __shfl_xor(v, 8, 32);
    acc[r] = v;
  }
  // lane 0 owns rows 0-7, lane 16 owns rows 8-15
  if (col == 0) {
#pragma unroll
    for (int r = 0; r < 8; ++r)
      atomicAdd(&out[brow[r]], 0.5f * acc[r]);
  }
}

extern "C" void kernel_launch(void* const* d_in, const int* in_sizes, int n_in,
                              void* d_out, int out_size, void* d_ws, size_t ws_size,
                              hipStream_t stream) {
  const float*     chg   = (const float*)d_in[0];      // charge_density [N,9] f32
  const float*     pos   = (const float*)d_in[1];      // positions [N,3] f32
  const long long* batch = (const long long*)d_in[2];  // batch [N] int64, sorted
  float* out = (float*)d_out;                          // energy [G] f32

  const int n   = in_sizes[2];            // 8192
  const int kch = in_sizes[0] / n;        // 9
  const int g   = out_size;               // 64

  zero_out_kernel<<<(g + 63) / 64, 64, 0, stream>>>(out, g);

  const int ntiles  = n / 16;             // 512 i-tiles, one per wave
  const int nblocks = (ntiles + 7) / 8;   // 8 waves (256 threads) per block
  electro_wmma_kernel<<<nblocks, 256, 0, stream>>>(chg, pos, batch, out, n, kch);
}